// FLAME_7164005449825
// MI455X (gfx1250) — compile-verified
//
#include <hip/hip_runtime.h>
#include <hip/hip_bf16.h>

typedef float v2f __attribute__((ext_vector_type(2)));
typedef float v8f __attribute__((ext_vector_type(8)));

#define BATCH 1024
#define NV    5023
#define N3    15069          // 3*NV
#define KBETA 400
#define KPOSE 36
#define NTILE ((N3 + 15) / 16)    // 942
#define NBLKX1 ((NTILE + 15) / 16) // 59  : 8 waves x 2 N-tiles per block
#define NBLKX2 ((NTILE + 7) / 8)   // 118 : 8 waves x 1 N-tile per block

// ---------------------------------------------------------------------------
// Kernel P: pack betas = concat(shape[300], expr[100]) -> (B,400)
// ---------------------------------------------------------------------------
__global__ void __launch_bounds__(256) pack_betas_kernel(
    const float* __restrict__ shape, const float* __restrict__ expr,
    float* __restrict__ betas) {
  int idx = blockIdx.x * 256 + threadIdx.x;
  if (idx >= BATCH * KBETA) return;
  int b = idx / KBETA, l = idx - b * KBETA;
  betas[idx] = (l < 300) ? shape[b * 300 + l] : expr[b * 100 + (l - 300)];
}

// ---------------------------------------------------------------------------
// Kernel A: v_shaped = v_template + betas @ shapedirs^T   (WMMA f32 16x16x4)
// Each wave owns TWO adjacent 16-col N-tiles sharing one A (betas) fragment:
//   per WMMA pair: 1 A b64 load + 2 B b64 loads  (A traffic halved)
// C accumulators preloaded with broadcast v_template row -> free epilogue.
// ---------------------------------------------------------------------------
__global__ void __launch_bounds__(256) gemm_shape_kernel(
    const float* __restrict__ betas,   // (B,400)
    const float* __restrict__ SD,      // shapedirs as (N3,400), K contiguous
    const float* __restrict__ vt,      // v_template flat (N3)
    float* __restrict__ out) {         // ws v_shaped (B,N3)
  const int lane = threadIdx.x & 31;
  const int wave = threadIdx.x >> 5;
  const int half = lane >> 4;          // 0: K pair 0/1, 1: K pair 2/3
  const int l16  = lane & 15;
  const int t0 = blockIdx.x * 16 + wave * 2;
  if (t0 >= NTILE) return;             // wave-uniform exit (EXEC stays full)
  const int b0 = blockIdx.y * 16;
  const int n0 = t0 * 16 + l16;
  const int n1 = n0 + 16;
  const int nc0 = (n0 < N3) ? n0 : (N3 - 1);   // clamp loads
  const int nc1 = (n1 < N3) ? n1 : (N3 - 1);   // clamp loads

  const float* ap  = betas + (size_t)(b0 + l16) * KBETA + 2 * half;
  const float* bp0 = SD + (size_t)nc0 * KBETA + 2 * half;
  const float* bp1 = SD + (size_t)nc1 * KBETA + 2 * half;
  __builtin_prefetch((const void*)bp0, 0, 1);
  __builtin_prefetch((const void*)bp1, 0, 1);
  __builtin_prefetch((const void*)ap, 0, 1);

  const float vt0 = vt[nc0];
  const float vt1 = vt[nc1];
  v8f acc0, acc1;
#pragma unroll
  for (int r = 0; r < 8; ++r) { acc0[r] = vt0; acc1[r] = vt1; }

#pragma unroll 2
  for (int k = 0; k < KBETA; k += 4) {
    v2f a  = *(const v2f*)ap;  ap  += 4;
    v2f b0 = *(const v2f*)bp0; bp0 += 4;
    v2f b1 = *(const v2f*)bp1; bp1 += 4;
    acc0 = __builtin_amdgcn_wmma_f32_16x16x4_f32(
        /*neg_a=*/false, a, /*neg_b=*/false, b0,
        /*c_mod=*/(short)0, acc0, /*reuse_a=*/false, /*reuse_b=*/false);
    acc1 = __builtin_amdgcn_wmma_f32_16x16x4_f32(
        false, a, false, b1, (short)0, acc1, false, false);
  }

  if (n0 < N3) {
#pragma unroll
    for (int r = 0; r < 8; ++r)
      out[(size_t)(b0 + r + 8 * half) * N3 + n0] = acc0[r];
  }
  if (n1 < N3) {
#pragma unroll
    for (int r = 0; r < 8; ++r)
      out[(size_t)(b0 + r + 8 * half) * N3 + n1] = acc1[r];
  }
}

// ---------------------------------------------------------------------------
// Kernel B: joints[b,j,c] = sum_v J_regressor[j,v] * v_shaped[b,v,c]
// one block per batch, LDS tree reduction of 15 partial sums
// ---------------------------------------------------------------------------
__global__ void __launch_bounds__(256) joints_kernel(
    const float* __restrict__ vsh, const float* __restrict__ JR,
    float* __restrict__ jout) {
  const int b = blockIdx.x;
  float acc[15];
#pragma unroll
  for (int q = 0; q < 15; ++q) acc[q] = 0.f;
  for (int v = threadIdx.x; v < NV; v += 256) {
    float x = vsh[(size_t)b * N3 + 3 * v + 0];
    float y = vsh[(size_t)b * N3 + 3 * v + 1];
    float z = vsh[(size_t)b * N3 + 3 * v + 2];
#pragma unroll
    for (int j = 0; j < 5; ++j) {
      float w = JR[j * NV + v];
      acc[j * 3 + 0] += w * x;
      acc[j * 3 + 1] += w * y;
      acc[j * 3 + 2] += w * z;
    }
  }
  __shared__ float s[256 * 15];
#pragma unroll
  for (int q = 0; q < 15; ++q) s[threadIdx.x * 15 + q] = acc[q];
  __syncthreads();
  for (int off = 128; off > 0; off >>= 1) {
    if ((int)threadIdx.x < off) {
#pragma unroll
      for (int q = 0; q < 15; ++q)
        s[threadIdx.x * 15 + q] += s[(threadIdx.x + off) * 15 + q];
    }
    __syncthreads();
  }
  if (threadIdx.x < 15) jout[b * 15 + threadIdx.x] = s[threadIdx.x];
}

// ---------------------------------------------------------------------------
// per-batch pose math helpers
// ---------------------------------------------------------------------------
__device__ inline void rot6d(const float* d, float* R) {
  float a1x = d[0], a1y = d[1], a1z = d[2];
  float a2x = d[3], a2y = d[4], a2z = d[5];
  float n1 = sqrtf(a1x * a1x + a1y * a1y + a1z * a1z);
  float b1x = a1x / n1, b1y = a1y / n1, b1z = a1z / n1;
  float dp = b1x * a2x + b1y * a2y + b1z * a2z;
  float b2x = a2x - dp * b1x, b2y = a2y - dp * b1y, b2z = a2z - dp * b1z;
  float n2 = sqrtf(b2x * b2x + b2y * b2y + b2z * b2z);
  b2x /= n2; b2y /= n2; b2z /= n2;
  float b3x = b1y * b2z - b1z * b2y;
  float b3y = b1z * b2x - b1x * b2z;
  float b3z = b1x * b2y - b1y * b2x;
  R[0] = b1x; R[1] = b1y; R[2] = b1z;
  R[3] = b2x; R[4] = b2y; R[5] = b2z;
  R[6] = b3x; R[7] = b3y; R[8] = b3z;
}

__device__ inline void matmul3(const float* A, const float* B, float* C) {
#pragma unroll
  for (int r = 0; r < 3; ++r)
#pragma unroll
    for (int c = 0; c < 3; ++c)
      C[r * 3 + c] = A[r * 3 + 0] * B[0 * 3 + c] +
                     A[r * 3 + 1] * B[1 * 3 + c] +
                     A[r * 3 + 2] * B[2 * 3 + c];
}

// ---------------------------------------------------------------------------
// Kernel C: per-batch rot mats, pose_feature, kinematic chain -> A_rel,
//           neck euler -> dynamic landmark row index y
// ---------------------------------------------------------------------------
__global__ void __launch_bounds__(256) pose_kernel(
    const float* __restrict__ rot, const float* __restrict__ neck,
    const float* __restrict__ jaw, const float* __restrict__ eye,
    const float* __restrict__ cams, const float* __restrict__ joints,
    float* __restrict__ pf, float* __restrict__ arel, int* __restrict__ yidx) {
  int b = blockIdx.x * 256 + threadIdx.x;
  if (b >= BATCH) return;

  float pose[30];
#pragma unroll
  for (int i = 0; i < 6; ++i) {
    pose[i]      = rot[b * 6 + i];
    pose[6 + i]  = neck[b * 6 + i];
    pose[12 + i] = jaw[b * 6 + i];
  }
#pragma unroll
  for (int i = 0; i < 12; ++i) pose[18 + i] = eye[b * 12 + i];

  float R[5][9];
#pragma unroll
  for (int j = 0; j < 5; ++j) rot6d(pose + j * 6, R[j]);

  // pose_feature = (R[1..4] - I).flatten
#pragma unroll
  for (int j = 1; j < 5; ++j)
#pragma unroll
    for (int e = 0; e < 9; ++e)
      pf[b * 36 + (j - 1) * 9 + e] = R[j][e] - ((e % 4 == 0) ? 1.f : 0.f);

  float J[5][3];
#pragma unroll
  for (int j = 0; j < 5; ++j)
#pragma unroll
    for (int c = 0; c < 3; ++c) J[j][c] = joints[b * 15 + j * 3 + c];

  const int par[5] = {-1, 0, 1, 1, 1};
  float rel[5][3];
#pragma unroll
  for (int c = 0; c < 3; ++c) rel[0][c] = J[0][c];
#pragma unroll
  for (int j = 1; j < 5; ++j)
#pragma unroll
    for (int c = 0; c < 3; ++c) rel[j][c] = J[j][c] - J[par[j]][c];

  float CR[5][9], CT[5][3];
#pragma unroll
  for (int e = 0; e < 9; ++e) CR[0][e] = R[0][e];
#pragma unroll
  for (int c = 0; c < 3; ++c) CT[0][c] = rel[0][c];
#pragma unroll
  for (int j = 1; j < 5; ++j) {
    int p = par[j];
    matmul3(CR[p], R[j], CR[j]);
#pragma unroll
    for (int r = 0; r < 3; ++r)
      CT[j][r] = CR[p][r * 3 + 0] * rel[j][0] + CR[p][r * 3 + 1] * rel[j][1] +
                 CR[p][r * 3 + 2] * rel[j][2] + CT[p][r];
  }

#pragma unroll
  for (int j = 0; j < 5; ++j) {
#pragma unroll
    for (int r = 0; r < 3; ++r) {
      float corr = CR[j][r * 3 + 0] * J[j][0] + CR[j][r * 3 + 1] * J[j][1] +
                   CR[j][r * 3 + 2] * J[j][2];
#pragma unroll
      for (int c = 0; c < 3; ++c)
        arel[b * 60 + j * 12 + r * 4 + c] = CR[j][r * 3 + c];
      arel[b * 60 + j * 12 + r * 4 + 3] = CT[j][r] - corr;
    }
  }

  // rel_rot = cameras @ (R_global @ R_neck)  (NECK_CHAIN = [1,0])
  float RR[9], M[9];
  matmul3(R[0], R[1], RR);
  matmul3(cams + (size_t)b * 9, RR, M);
  float sy = sqrtf(M[0] * M[0] + M[3] * M[3]);
  float euler = atan2f(-M[6], sy);
  float deg = -euler * 57.29577951308232f;
  int y = (int)rintf(fminf(deg, 39.0f));
  if (y < 0) y = (y < -39) ? 78 : (39 - y);
  yidx[b] = y;
}

// ---------------------------------------------------------------------------
// Kernel D: v_posed = v_shaped + pose_feature @ posedirs (WMMA, C from ws,
//           in-place update of ws).  K = 36 = 9 WMMA steps.
// ---------------------------------------------------------------------------
__global__ void __launch_bounds__(256) gemm_pose_kernel(
    const float* __restrict__ pf,     // (B,36)
    const float* __restrict__ PD,     // posedirs (36,N3), n contiguous
    float* __restrict__ vw) {         // ws v_shaped -> v_posed in place
  const int lane = threadIdx.x & 31;
  const int wave = threadIdx.x >> 5;
  const int half = lane >> 4;
  const int l16  = lane & 15;
  const int ntile = blockIdx.x * 8 + wave;
  if (ntile >= NTILE) return;
  const int b0 = blockIdx.y * 16;
  const int n  = ntile * 16 + l16;
  const int nc = (n < N3) ? n : (N3 - 1);

  const float* ap = pf + (size_t)(b0 + l16) * KPOSE + 2 * half;

  v8f acc;
#pragma unroll
  for (int r = 0; r < 8; ++r)
    acc[r] = vw[(size_t)(b0 + r + 8 * half) * N3 + nc];

#pragma unroll
  for (int k = 0; k < KPOSE; k += 4) {
    v2f a = *(const v2f*)ap; ap += 4;
    v2f b;
    b.x = PD[(size_t)(k + 2 * half + 0) * N3 + nc];
    b.y = PD[(size_t)(k + 2 * half + 1) * N3 + nc];
    acc = __builtin_amdgcn_wmma_f32_16x16x4_f32(
        false, a, false, b, (short)0, acc, false, false);
  }

  if (n < N3) {
#pragma unroll
    for (int r = 0; r < 8; ++r)
      vw[(size_t)(b0 + r + 8 * half) * N3 + n] = acc[r];
  }
}

// ---------------------------------------------------------------------------
// Kernel E: LBS blend + eyelids + translation -> verts (d_out)
// one block = 256 vertices of one batch; A_rel broadcast via LDS
// ---------------------------------------------------------------------------
__global__ void __launch_bounds__(256) lbs_kernel(
    const float* __restrict__ vw,     // v_posed (B,N3)
    const float* __restrict__ arel,   // (B,5,12)
    const float* __restrict__ lw,     // lbs_weights (V,5)
    const float* __restrict__ leye, const float* __restrict__ reye,  // (V,3)
    const float* __restrict__ eyep,   // (B,2)
    const float* __restrict__ trans,  // (B,3)
    float* __restrict__ verts) {      // d_out (B,V,3)
  const int b = blockIdx.y;
  const int v = blockIdx.x * 256 + threadIdx.x;
  __shared__ float A[60];
  __shared__ float tr[3];
  __shared__ float ey[2];
  if (threadIdx.x < 60) A[threadIdx.x] = arel[b * 60 + threadIdx.x];
  else if (threadIdx.x < 63) tr[threadIdx.x - 60] = trans[b * 3 + threadIdx.x - 60];
  else if (threadIdx.x < 65) ey[threadIdx.x - 63] = eyep[b * 2 + threadIdx.x - 63];
  __syncthreads();
  if (v >= NV) return;

  float w[5];
#pragma unroll
  for (int j = 0; j < 5; ++j) w[j] = lw[v * 5 + j];
  float T[12];
#pragma unroll
  for (int e = 0; e < 12; ++e) {
    float t = 0.f;
#pragma unroll
    for (int j = 0; j < 5; ++j) t += w[j] * A[j * 12 + e];
    T[e] = t;
  }
  float p0 = vw[(size_t)b * N3 + 3 * v + 0];
  float p1 = vw[(size_t)b * N3 + 3 * v + 1];
  float p2 = vw[(size_t)b * N3 + 3 * v + 2];
#pragma unroll
  for (int r = 0; r < 3; ++r) {
    float o = T[r * 4 + 0] * p0 + T[r * 4 + 1] * p1 + T[r * 4 + 2] * p2 +
              T[r * 4 + 3];
    o += reye[v * 3 + r] * ey[1] + leye[v * 3 + r] * ey[0] + tr[r];
    verts[(size_t)b * N3 + 3 * v + r] = o;
  }
}

// ---------------------------------------------------------------------------
// Kernel F: landmarks (17 dynamic + 51 static) and mediapipe (105)
// ---------------------------------------------------------------------------
__global__ void __launch_bounds__(192) lmk_kernel(
    const float* __restrict__ verts,  // d_out verts
    const int* __restrict__ faces,    // (NF,3)
    const int* __restrict__ yidx,     // (B,)
    const int* __restrict__ dynfi,    // (79,17)
    const float* __restrict__ dynbc,  // (79,17,3)
    const int* __restrict__ statfi,   // (51,)
    const float* __restrict__ statbc, // (51,3)
    const int* __restrict__ mpfi,     // (105,)
    const float* __restrict__ mpbc,   // (105,3)
    float* __restrict__ lmk_out,      // (B,68,3)
    float* __restrict__ mp_out) {     // (B,105,3)
  const int b = blockIdx.x;
  const int t = threadIdx.x;
  int fi;
  float bc0, bc1, bc2;
  float* dst;
  if (t < 68) {
    if (t < 17) {
      int y = yidx[b];
      fi  = dynfi[y * 17 + t];
      bc0 = dynbc[(y * 17 + t) * 3 + 0];
      bc1 = dynbc[(y * 17 + t) * 3 + 1];
      bc2 = dynbc[(y * 17 + t) * 3 + 2];
    } else {
      fi  = statfi[t - 17];
      bc0 = statbc[(t - 17) * 3 + 0];
      bc1 = statbc[(t - 17) * 3 + 1];
      bc2 = statbc[(t - 17) * 3 + 2];
    }
    dst = lmk_out + ((size_t)b * 68 + t) * 3;
  } else if (t < 173) {
    int q = t - 68;
    fi  = mpfi[q];
    bc0 = mpbc[q * 3 + 0];
    bc1 = mpbc[q * 3 + 1];
    bc2 = mpbc[q * 3 + 2];
    dst = mp_out + ((size_t)b * 105 + q) * 3;
  } else {
    return;
  }
  int v0 = faces[fi * 3 + 0], v1 = faces[fi * 3 + 1], v2 = faces[fi * 3 + 2];
  const float* vb = verts + (size_t)b * N3;
#pragma unroll
  for (int i = 0; i < 3; ++i)
    dst[i] = bc0 * vb[v0 * 3 + i] + bc1 * vb[v1 * 3 + i] + bc2 * vb[v2 * 3 + i];
}

// ---------------------------------------------------------------------------
extern "C" void kernel_launch(void* const* d_in, const int* in_sizes, int n_in,
                              void* d_out, int out_size, void* d_ws, size_t ws_size,
                              hipStream_t stream) {
  (void)in_sizes; (void)n_in; (void)out_size; (void)ws_size;
  const float* shape  = (const float*)d_in[0];
  const float* expr   = (const float*)d_in[1];
  const float* trans  = (const float*)d_in[2];
  const float* rot    = (const float*)d_in[3];
  const float* neck   = (const float*)d_in[4];
  const float* jaw    = (const float*)d_in[5];
  const float* eye    = (const float*)d_in[6];
  const float* eyep   = (const float*)d_in[7];
  const float* cams   = (const float*)d_in[8];
  const float* vtmpl  = (const float*)d_in[9];
  const float* sdirs  = (const float*)d_in[10];
  const float* pdirs  = (const float*)d_in[11];
  const float* jreg   = (const float*)d_in[12];
  const float* lbsw   = (const float*)d_in[13];
  const float* leye   = (const float*)d_in[14];
  const float* reye   = (const float*)d_in[15];
  const float* statbc = (const float*)d_in[16];
  const float* dynbc  = (const float*)d_in[17];
  const float* mpbc   = (const float*)d_in[18];
  const int*   faces  = (const int*)d_in[19];
  const int*   statfi = (const int*)d_in[20];
  const int*   dynfi  = (const int*)d_in[21];
  const int*   mpfi   = (const int*)d_in[22];

  float* ws = (float*)d_ws;
  size_t off = 0;
  float* ws_v      = ws + off; off += (size_t)BATCH * N3;     // 15,430,656
  float* ws_betas  = ws + off; off += (size_t)BATCH * KBETA;  // 409,600
  float* ws_joints = ws + off; off += (size_t)BATCH * 15;
  float* ws_pf     = ws + off; off += (size_t)BATCH * KPOSE;
  float* ws_arel   = ws + off; off += (size_t)BATCH * 60;
  int*   ws_y      = (int*)(ws + off);

  float* out_verts = (float*)d_out;
  float* out_lmk   = out_verts + (size_t)BATCH * N3;
  float* out_mp    = out_lmk + (size_t)BATCH * 68 * 3;

  pack_betas_kernel<<<(BATCH * KBETA + 255) / 256, 256, 0, stream>>>(
      shape, expr, ws_betas);
  gemm_shape_kernel<<<dim3(NBLKX1, BATCH / 16), 256, 0, stream>>>(
      ws_betas, sdirs, vtmpl, ws_v);
  joints_kernel<<<BATCH, 256, 0, stream>>>(ws_v, jreg, ws_joints);
  pose_kernel<<<(BATCH + 255) / 256, 256, 0, stream>>>(
      rot, neck, jaw, eye, cams, ws_joints, ws_pf, ws_arel, ws_y);
  gemm_pose_kernel<<<dim3(NBLKX2, BATCH / 16), 256, 0, stream>>>(
      ws_pf, pdirs, ws_v);
  lbs_kernel<<<dim3((NV + 255) / 256, BATCH), 256, 0, stream>>>(
      ws_v, ws_arel, lbsw, leye, reye, eyep, trans, out_verts);
  lmk_kernel<<<BATCH, 192, 0, stream>>>(
      out_verts, faces, ws_y, dynfi, dynbc, statfi, statbc, mpfi, mpbc,
      out_lmk, out_mp);
}